// HetGCN_44229573214371
// MI455X (gfx1250) — compile-verified
//
#include <hip/hip_runtime.h>
#include <math.h>
#include <stdint.h>

typedef __attribute__((ext_vector_type(2))) float v2f;
typedef __attribute__((ext_vector_type(4))) float v4f;
typedef __attribute__((ext_vector_type(8))) float v8f;
typedef int v4i __attribute__((vector_size(16)));   // matches builtin's pointee type

#define N_NODES 8192
#define D 32

// ---- gfx1250 async global->LDS path (guarded; fallback = sync staging) ----
#if defined(__has_builtin)
#if __has_builtin(__builtin_amdgcn_global_load_async_to_lds_b128) && \
    __has_builtin(__builtin_amdgcn_s_wait_asynccnt)
#define USE_ASYNC 1
#endif
#endif
#ifndef USE_ASYNC
#define USE_ASYNC 0
#endif

#if USE_ASYNC
// generic->AS3 is low-32-bit truncation;
// generic->AS1 is representation-identical. Integer casts sidestep addrspace-cast rules.
// Builtin signature (from diagnostics): (v4i AS1*, v4i AS3*, imm offset, imm cpol).
#define ASYNC_B128(gp, lp)                                                        \
    __builtin_amdgcn_global_load_async_to_lds_b128(                               \
        (__attribute__((address_space(1))) v4i*)(uintptr_t)(gp),                  \
        (__attribute__((address_space(3))) v4i*)(uint32_t)(uintptr_t)(lp), 0, 0)
#endif

// ---------------- degree vectors ----------------
__global__ __launch_bounds__(256) void rowsum_rsqrt_kernel(const float* __restrict__ A,
                                                           float addend,
                                                           float* __restrict__ out) {
    __shared__ float red[256];
    const int row = blockIdx.x;
    const v4f* Ar = (const v4f*)(A + (size_t)row * N_NODES);
    float s = 0.f;
    for (int j = threadIdx.x; j < N_NODES / 4; j += 256) {
        v4f v = __builtin_nontemporal_load(Ar + j);
        s += (v.x + v.y) + (v.z + v.w);
    }
    red[threadIdx.x] = s;
    __syncthreads();
    for (int step = 128; step > 0; step >>= 1) {
        if (threadIdx.x < step) red[threadIdx.x] += red[threadIdx.x + step];
        __syncthreads();
    }
    if (threadIdx.x == 0) out[row] = rsqrtf(addend + red[0]);
}

__global__ __launch_bounds__(256) void colsum_rsqrt_kernel(const float* __restrict__ A,
                                                           float* __restrict__ out) {
    const int col = blockIdx.x * 256 + threadIdx.x;
    float s0 = 0.f, s1 = 0.f, s2 = 0.f, s3 = 0.f;
    for (int i = 0; i < N_NODES; i += 4) {
        s0 += __builtin_nontemporal_load(A + (size_t)(i + 0) * N_NODES + col);
        s1 += __builtin_nontemporal_load(A + (size_t)(i + 1) * N_NODES + col);
        s2 += __builtin_nontemporal_load(A + (size_t)(i + 2) * N_NODES + col);
        s3 += __builtin_nontemporal_load(A + (size_t)(i + 3) * N_NODES + col);
    }
    out[col] = rsqrtf((s0 + s1) + (s2 + s3));
}

// ---------------- tiny 32x32 @ 32x32 (proj_u @ weight_c) ----------------
__global__ void mm32_kernel(const float* __restrict__ P, const float* __restrict__ W,
                            float* __restrict__ M) {
    const int i = threadIdx.x >> 5;
    const int j = threadIdx.x & 31;
    float s = 0.f;
#pragma unroll
    for (int k = 0; k < 32; ++k) s += P[i * 32 + k] * W[k * 32 + j];
    M[i * 32 + j] = s;
}

// ---------------- support = colscale[row] * (X @ W) ----------------
__global__ __launch_bounds__(256) void support_kernel(const float* __restrict__ X,
                                                      const float* __restrict__ W,
                                                      const float* __restrict__ scale,
                                                      float* __restrict__ out) {
    __shared__ float Ws[32][33];
    const int t = threadIdx.x;
    for (int i = t; i < 32 * 32; i += 256) Ws[i >> 5][i & 31] = W[i];
    __syncthreads();
    const int row = blockIdx.x * 8 + (t >> 5);
    const int col = t & 31;
    const float* Xr = X + (size_t)row * D;
    float s = 0.f;
#pragma unroll
    for (int k = 0; k < 32; ++k) s += Xr[k] * Ws[k][col];
    out[(size_t)row * D + col] = scale[row] * s;
}

// ---------------- main: Y = rowscale .* (A @ S), fp32 WMMA ----------------
// A: [8192 x 8192], S: [8192 x 32] (pre-scaled by column-side degree), Y: [8192 x 32]
struct MMArgs {
    const float* A[3];
    const float* S[3];
    const float* rs[3];
    float* Y[3];
};

__global__ __launch_bounds__(256) void rowscale_mm_kernel(MMArgs args) {
    const int pb = blockIdx.y;
    const float* __restrict__ A  = args.A[pb];
    const float* __restrict__ S  = args.S[pb];
    const float* __restrict__ rs = args.rs[pb];
    float* __restrict__ Y        = args.Y[pb];

    // pads chosen for conflict-free ds reads:
    //  A rows stride 68 -> lanes hit banks {4m..4m+3} (disjoint across the wave)
    //  S rows stride 40 -> hi/lo half-wave bank sets disjoint
    __shared__ float As[2][64][68];
    __shared__ float Ss[2][64][40];

    const int t = threadIdx.x;
    const int rowBase = blockIdx.x * 64;

    const int wave = t >> 5;            // 8 waves
    const int lane = t & 31;
    const int m0 = (wave >> 1) * 16;    // row tile within 64
    const int n0 = (wave & 1) * 16;     // col tile within 32
    const int ml = lane & 15;
    const int hi = lane >> 4;           // half-wave selects K pair {0,1} vs {2,3}

    // cooperative-load coordinates
    const int ar = t >> 4;              // 0..15
    const int ac = (t & 15) * 4;        // 0..60
    const int sr = t >> 3;              // 0..31
    const int sc = (t & 7) * 4;         // 0..28

    v8f acc = {};

#if USE_ASYNC
    // ---- double-buffered async pipeline: 6 async b128 ops per tile ----
    {
        // prologue: tile 0 into buffer 0
        #pragma unroll
        for (int rr = 0; rr < 64; rr += 16)
            ASYNC_B128(A + (size_t)(rowBase + ar + rr) * N_NODES + ac, &As[0][ar + rr][ac]);
        ASYNC_B128(S + (size_t)sr * D + sc,        &Ss[0][sr][sc]);
        ASYNC_B128(S + (size_t)(sr + 32) * D + sc, &Ss[0][sr + 32][sc]);
    }
    int buf = 0;
    for (int kb = 0; kb < N_NODES; kb += 64, buf ^= 1) {
        const bool hasNext = (kb + 64) < N_NODES;
        if (hasNext) {
            const int kn = kb + 64;
            #pragma unroll
            for (int rr = 0; rr < 64; rr += 16)
                ASYNC_B128(A + (size_t)(rowBase + ar + rr) * N_NODES + (kn + ac),
                           &As[buf ^ 1][ar + rr][ac]);
            ASYNC_B128(S + (size_t)(kn + sr) * D + sc,        &Ss[buf ^ 1][sr][sc]);
            ASYNC_B128(S + (size_t)(kn + sr + 32) * D + sc,   &Ss[buf ^ 1][sr + 32][sc]);
            __builtin_amdgcn_s_wait_asynccnt(6);   // current tile's 6 ops landed
        } else {
            __builtin_amdgcn_s_wait_asynccnt(0);
        }
        __syncthreads();                            // everyone's tile visible in LDS

        const float (*Asb)[68] = As[buf];
        const float (*Ssb)[40] = Ss[buf];
        #pragma unroll
        for (int kk = 0; kk < 64; kk += 4) {
            const int kb2 = kk + 2 * hi;
            v2f a = *(const v2f*)&Asb[m0 + ml][kb2];
            v2f b;
            b.x = Ssb[kb2][n0 + ml];
            b.y = Ssb[kb2 + 1][n0 + ml];
            acc = __builtin_amdgcn_wmma_f32_16x16x4_f32(
                false, a, false, b, (short)0, acc, false, false);
        }
        __syncthreads();                            // safe to overwrite this buffer
    }
#else
    // ---- fallback: synchronous staging (known-good) ----
    for (int kb = 0; kb < N_NODES; kb += 64) {
        #pragma unroll
        for (int rr = 0; rr < 64; rr += 16) {
            const float* gp = A + (size_t)(rowBase + ar + rr) * N_NODES + (kb + ac);
            *(v4f*)&As[0][ar + rr][ac] = __builtin_nontemporal_load((const v4f*)gp);
            if (kb + 64 < N_NODES) __builtin_prefetch(gp + 64, 0, 0);
        }
        *(v4f*)&Ss[0][sr][sc]      = *(const v4f*)(S + (size_t)(kb + sr) * D + sc);
        *(v4f*)&Ss[0][sr + 32][sc] = *(const v4f*)(S + (size_t)(kb + sr + 32) * D + sc);
        __syncthreads();
        #pragma unroll
        for (int kk = 0; kk < 64; kk += 4) {
            const int kb2 = kk + 2 * hi;
            v2f a = *(const v2f*)&As[0][m0 + ml][kb2];
            v2f b;
            b.x = Ss[0][kb2][n0 + ml];
            b.y = Ss[0][kb2 + 1][n0 + ml];
            acc = __builtin_amdgcn_wmma_f32_16x16x4_f32(
                false, a, false, b, (short)0, acc, false, false);
        }
        __syncthreads();
    }
#endif

    // C/D layout: VGPR v -> (M = v + 8*hi, N = ml) within the 16x16 tile.
#pragma unroll
    for (int v = 0; v < 8; ++v) {
        const int row = rowBase + m0 + v + 8 * hi;
        Y[(size_t)row * D + (n0 + ml)] = rs[row] * acc[v];
    }
}

// ---------------- attention epilogue ----------------
__global__ __launch_bounds__(256) void epilogue_kernel(const float* __restrict__ E1,
                                                       const float* __restrict__ E2,
                                                       const float* __restrict__ XC,
                                                       const float* __restrict__ Wlin,  // [64x32]
                                                       const float* __restrict__ avec,  // [32]
                                                       float* __restrict__ outC,
                                                       float* __restrict__ outYc,
                                                       float* __restrict__ outYu) {
    __shared__ float Wl[64][33];
    __shared__ float av[32];
    const int t = threadIdx.x;
    for (int i = t; i < 64 * 32; i += 256) Wl[i >> 5][i & 31] = Wlin[i];
    if (t < 32) av[t] = avec[t];
    __syncthreads();

    const int lane = t & 31;
    const int wave = t >> 5;
    const int row = blockIdx.x * 8 + wave;  // grid 1024 covers 8192 exactly

    const size_t idx = (size_t)row * D + lane;
    const float e1v = E1[idx];
    const float e2v = E2[idx];
    const float xcv = XC[idx];

    // t[n] = sum_k cat[k] * Wlin[k][n], cat = [E, X_C]; lane owns column n = lane.
    float s1 = 0.f, s2 = 0.f;
#pragma unroll
    for (int k = 0; k < 32; ++k) {
        const float w0 = Wl[k][lane];
        const float w1 = Wl[k + 32][lane];
        const float b1 = __shfl(e1v, k, 32);
        const float b2 = __shfl(e2v, k, 32);
        const float bx = __shfl(xcv, k, 32);
        s1 += b1 * w0 + bx * w1;
        s2 += b2 * w0 + bx * w1;
    }
    s1 = fmaxf(s1, 0.f) * av[lane];
    s2 = fmaxf(s2, 0.f) * av[lane];
#pragma unroll
    for (int off = 16; off > 0; off >>= 1) {
        s1 += __shfl_xor(s1, off, 32);
        s2 += __shfl_xor(s2, off, 32);
    }
    const float m   = fmaxf(s1, s2);
    const float w1e = __expf(s1 - m);
    const float w2e = __expf(s2 - m);
    const float inv = 1.f / (w1e + w2e);
    const float yc = (w1e * inv) * e1v;
    const float yu = (w2e * inv) * e2v;
    outYc[idx] = yc;
    outYu[idx] = yu;
    outC[idx]  = yc + yu;
}

extern "C" void kernel_launch(void* const* d_in, const int* in_sizes, int n_in,
                              void* d_out, int out_size, void* d_ws, size_t ws_size,
                              hipStream_t stream) {
    (void)in_sizes; (void)n_in; (void)out_size; (void)ws_size;
    const float* X_U      = (const float*)d_in[0];
    const float* X_C      = (const float*)d_in[1];
    const float* adj_uu   = (const float*)d_in[2];
    const float* adj_uc   = (const float*)d_in[3];
    const float* adj_cc   = (const float*)d_in[4];
    const float* weight_s = (const float*)d_in[5];
    const float* proj_u   = (const float*)d_in[6];
    /* proj_c (d_in[7]) unused in the forward math */
    const float* weight_c = (const float*)d_in[8];
    const float* W_lin    = (const float*)d_in[9];
    const float* a_vec    = (const float*)d_in[10];

    float* ws      = (float*)d_ws;
    float* dinv_uu = ws;                   // 8192
    float* dinv_cc = ws + 8192;            // 8192
    float* du      = ws + 16384;           // 8192
    float* dc      = ws + 24576;           // 8192
    float* Mpc     = ws + 32768;           // 1024 (proj_u @ weight_c)
    float* sup_uu  = ws + 33792;           // 262144
    float* sup_cc  = sup_uu + 262144;      // 262144
    float* sup_uc  = sup_cc + 262144;      // 262144
    float* E1      = sup_uc + 262144;      // 262144 (node_embed_c1)
    float* E2      = E1 + 262144;          // 262144 (node_embed_c2)

    float* out_u  = (float*)d_out;         // node_embed_u
    float* out_c  = out_u + 262144;        // node_embed_c
    float* out_yc = out_c + 262144;        // Y_c
    float* out_yu = out_yc + 262144;       // Y_u

    rowsum_rsqrt_kernel<<<N_NODES, 256, 0, stream>>>(adj_uu, 1.0f, dinv_uu);
    rowsum_rsqrt_kernel<<<N_NODES, 256, 0, stream>>>(adj_cc, 1.0f, dinv_cc);
    rowsum_rsqrt_kernel<<<N_NODES, 256, 0, stream>>>(adj_uc, 0.0f, du);
    colsum_rsqrt_kernel<<<N_NODES / 256, 256, 0, stream>>>(adj_uc, dc);

    mm32_kernel<<<1, 1024, 0, stream>>>(proj_u, weight_c, Mpc);

    support_kernel<<<N_NODES / 8, 256, 0, stream>>>(X_U, weight_s, dinv_uu, sup_uu);
    support_kernel<<<N_NODES / 8, 256, 0, stream>>>(X_C, weight_s, dinv_cc, sup_cc);
    support_kernel<<<N_NODES / 8, 256, 0, stream>>>(X_U, Mpc, dc, sup_uc);

    MMArgs ma;
    ma.A[0] = adj_uu; ma.S[0] = sup_uu; ma.rs[0] = dinv_uu; ma.Y[0] = out_u;
    ma.A[1] = adj_cc; ma.S[1] = sup_cc; ma.rs[1] = dinv_cc; ma.Y[1] = E1;
    ma.A[2] = adj_uc; ma.S[2] = sup_uc; ma.rs[2] = du;      ma.Y[2] = E2;
    rowscale_mm_kernel<<<dim3(N_NODES / 64, 3), 256, 0, stream>>>(ma);

    epilogue_kernel<<<N_NODES / 8, 256, 0, stream>>>(E1, E2, X_C, W_lin, a_vec,
                                                     out_c, out_yc, out_yu);
}